// SelfCorrelationComputation_47888885350410
// MI455X (gfx1250) — compile-verified
//
#include <hip/hip_runtime.h>

// Problem constants (from reference): x[8,64,80,80] f32, 5x5 windows, pad 2.
#define B_      8
#define C_      64
#define H_      80
#define W_      80
#define PLANE   (H_ * W_)            // 6400 elements per (b,c) plane
#define KK      5
#define WIN     (KK * KK)            // 25
#define OUTPP   (PLANE * WIN)        // 160000 outputs per plane
#define NPLANES (B_ * C_)            // 512
#define EPS_F   1e-12f

// Zero-padded LDS plane geometry: 2 pad rows top/bottom, 4 pad cols left/right
// so every interior row starts 16B-aligned (88 floats * 4B = 352B stride).
#define PSTR    88
#define PROWS   84
#define PSIZE   (PROWS * PSTR)       // 7392 floats = 29.6 KB
#define PC0     (2 * PSTR + 4)       // padded-coords offset of pixel (0,0)

// Types matching the async builtin's parameters: (v4i AS1*, v4i AS3*, Ii, Ii)
typedef int v4i_t __attribute__((vector_size(16)));
typedef __attribute__((address_space(1))) v4i_t gas_v4i;
typedef __attribute__((address_space(3))) v4i_t las_v4i;

// ---- CDNA5 async global -> LDS copy (ASYNCcnt path) ------------------------
__device__ __forceinline__ void async_copy_b128_to_lds(const void* gptr, void* lptr) {
#if __has_builtin(__builtin_amdgcn_global_load_async_to_lds_b128)
    __builtin_amdgcn_global_load_async_to_lds_b128(
        (gas_v4i*)gptr, (las_v4i*)lptr, /*offset=*/0, /*cpol=*/0);
#else
    unsigned lds_off = (unsigned)(unsigned long long)lptr;  // low 32 bits == LDS offset
    asm volatile("global_load_async_to_lds_b128 %0, %1, off"
                 :
                 : "v"(lds_off), "v"(gptr)
                 : "memory");
#endif
}

__device__ __forceinline__ void wait_async_zero() {
#if __has_builtin(__builtin_amdgcn_s_wait_asynccnt)
    __builtin_amdgcn_s_wait_asynccnt(0);
#else
    asm volatile("s_wait_asynccnt 0" ::: "memory");
#endif
}

// ---- Kernel 2: windowed self-correlation, one block per (b,c) plane --------
__global__ __launch_bounds__(256) void corr_kernel(const float* __restrict__ x,
                                                   const float* __restrict__ invn,
                                                   float* __restrict__ out) {
    __shared__ alignas(16) float sp[PSIZE];   // zero-padded normalized plane
    __shared__ alignas(16) float sn[PLANE];   // inverse norms for this batch
    __shared__ int lut[WIN];                  // ij -> neighbor delta in padded coords

    const unsigned tid   = threadIdx.x;
    const unsigned plane = blockIdx.x;        // [0, 512)
    const unsigned b     = plane >> 6;        // plane / C_
    const float* gx = x    + (size_t)plane * PLANE;
    const float* gn = invn + (size_t)b     * PLANE;

    // 1) Zero the padded plane (borders matter; interior is overwritten below).
    for (unsigned i = tid; i < PSIZE / 4; i += 256u) {
        ((float4*)sp)[i] = make_float4(0.f, 0.f, 0.f, 0.f);
    }
    if (tid < WIN) {
        int ki = (int)tid / KK, kj = (int)tid - ki * KK;
        lut[tid] = (ki - 2) * PSTR + (kj - 2);
    }
    __syncthreads();  // DS zeroes complete before the async engine writes interior

    // 2) Async-stage x rows into the padded interior (20 b128 chunks / row)
    //    and the inv-norm plane, via the CDNA5 async global->LDS path.
    for (unsigned t = tid; t < (unsigned)(H_ * W_ / 4); t += 256u) {  // 1600 chunks
        unsigned h = t / (unsigned)(W_ / 4);                          // /20
        unsigned k = t - h * (unsigned)(W_ / 4);
        async_copy_b128_to_lds(gx + h * W_ + 4u * k,
                               &sp[(h + 2u) * PSTR + 4u + 4u * k]);   // 16B aligned
    }
    for (unsigned i = tid; i < (unsigned)(PLANE / 4); i += 256u) {
        async_copy_b128_to_lds(gn + 4u * i, &sn[4u * i]);
    }
    wait_async_zero();
    __syncthreads();

    // 3) Normalize interior in place: xn = relu(x) * invnorm.
    for (unsigned i = tid; i < (unsigned)PLANE; i += 256u) {
        unsigned h = i / (unsigned)W_;
        unsigned p = PC0 + i + 8u * h;        // (h+2)*88 + (i%80) + 4
        sp[p] = fmaxf(sp[p], 0.0f) * sn[i];
    }
    __syncthreads();

    // 4) Main loop: branch-free LDS gathers, fully coalesced b128 stores.
    //    Flat output index within plane: o = pix*25 + ij.
    float* op = out + (size_t)plane * OUTPP;
    for (unsigned o4 = tid; o4 < (unsigned)(OUTPP / 4); o4 += 256u) {
        float4 r;
        float* re = (float*)&r;
        unsigned o = o4 * 4u;
#pragma unroll
        for (int e = 0; e < 4; ++e) {
            unsigned oe  = o + (unsigned)e;
            unsigned pix = oe / 25u;
            unsigned ij  = oe - pix * 25u;
            unsigned h   = pix / (unsigned)W_;
            unsigned c   = PC0 + pix + 8u * h;          // padded center index
            re[e] = sp[c] * sp[(int)c + lut[ij]];       // zero pad => no bounds check
        }
        ((float4*)op)[o4] = r;   // 16B aligned: plane*OUTPP*4 and o*4 are 16B multiples
    }
}

// ---- Kernel 1: inverse L3 norm per (b,h,w) ---------------------------------
// invn[b,h,w] = 1 / max(cbrt(sum_c relu(x)^3), eps)
__global__ __launch_bounds__(256) void norm_kernel(const float* __restrict__ x,
                                                   float* __restrict__ invn) {
    unsigned idx = blockIdx.x * blockDim.x + threadIdx.x;   // [0, B_*PLANE)
    if (idx >= (unsigned)(B_ * PLANE)) return;
    unsigned b = idx / (unsigned)PLANE;
    unsigned p = idx - b * (unsigned)PLANE;
    const float* base = x + (size_t)b * C_ * PLANE + p;
    float s = 0.0f;
#pragma unroll 8
    for (int c = 0; c < C_; ++c) {
        float v = fmaxf(base[(size_t)c * PLANE], 0.0f);
        s = fmaf(v * v, v, s);                               // v >= 0 so |v|^3 == v^3
    }
    float n = fmaxf(cbrtf(s), EPS_F);
    invn[idx] = 1.0f / n;
}

extern "C" void kernel_launch(void* const* d_in, const int* in_sizes, int n_in,
                              void* d_out, int out_size, void* d_ws, size_t ws_size,
                              hipStream_t stream) {
    (void)in_sizes; (void)n_in; (void)out_size; (void)ws_size;
    const float* x    = (const float*)d_in[0];
    float*       out  = (float*)d_out;
    float*       invn = (float*)d_ws;          // needs 8*6400*4 = 204800 bytes

    const int threads = 256;
    const int gnorm = (B_ * PLANE + threads - 1) / threads;  // 200 blocks
    norm_kernel<<<gnorm, threads, 0, stream>>>(x, invn);
    corr_kernel<<<NPLANES, threads, 0, stream>>>(x, invn, out);
}